// ModulatedDeformConv_27178553049518
// MI455X (gfx1250) — compile-verified
//
#include <hip/hip_runtime.h>

typedef __attribute__((ext_vector_type(2))) float v2f;
typedef __attribute__((ext_vector_type(8))) float v8f;

#define B_   8
#define C_   256
#define CO_  256
#define KT_  2304   /* C*9 */
#define HW_  4096   /* 64*64 */
#define ASTR 20     /* A-tile LDS row stride (words); 80B pitch -> 16B aligned segs */
#define BP_  130    /* B-tile LDS pitch per k-pair (words): 64*2 + 2 pad */
#define NCHUNK (KT_ / 16)

// Async DMA: copy 16 bytes from global to LDS, tracked by ASYNCcnt.
__device__ __forceinline__ void async_lds_b128(unsigned lds_byte_off, const float* gsrc) {
    asm volatile("global_load_async_to_lds_b128 %0, %1, off"
                 :: "v"(lds_byte_off), "v"(gsrc)
                 : "memory");
}
__device__ __forceinline__ void wait_async0() {
    asm volatile("s_wait_asynccnt 0" ::: "memory");
}

// Fused deformable-im2col + WMMA f32 GEMM, double-buffered LDS, async weight DMA.
// Workgroup: 256 threads (8 waves) computes the full M=256 (all Co) x N=64
// (one output row p) tile for one batch image.
__global__ __launch_bounds__(256) void dcn_wmma_kernel(
    const float* __restrict__ x,      // [8,256,64,64]
    const float* __restrict__ offset, // [8,18,64,64]
    const float* __restrict__ mask,   // [8,9,64,64]
    const float* __restrict__ weight, // [256,256,3,3] == [256,2304]
    const float* __restrict__ bias,   // [256]
    float* __restrict__ out)          // [8,256,64,64]
{
    __shared__ unsigned short s_idx[576 * 4];     // per (tap,q): 4 clamped corner indices
    __shared__ float          s_wgt[576 * 4];     // per (tap,q): bilinear wgts * mask (0 if OOB)
    __shared__ float          s_A[2][256 * ASTR]; // weight tiles (double buffered)
    __shared__ float          s_B[2][8 * BP_];    // col tiles, k-pair interleaved (dbl buf)

    const int t = threadIdx.x;
    const int b = blockIdx.x >> 6;     // batch
    const int p = blockIdx.x & 63;     // output row; pixels n = p*64 + q

    // ---------- precompute bilinear taps for this output row ----------
    for (int e = t; e < 576; e += 256) {
        const int k = e >> 6;          // tap 0..8
        const int q = e & 63;          // output column
        const float dy = offset[(((b * 18) + 2 * k    ) * 64 + p) * 64 + q];
        const float dx = offset[(((b * 18) + 2 * k + 1) * 64 + p) * 64 + q];
        const float mv = mask  [(((b * 9 ) +     k    ) * 64 + p) * 64 + q];
        const float py = dy + (float)(p - 1 + (k / 3));
        const float px = dx + (float)(q - 1 + (k % 3));
        const float fy = floorf(py);
        const float fx = floorf(px);
        const int   y0 = (int)fy;
        const int   x0 = (int)fx;
        const float ly = py - fy;
        const float lx = px - fx;
        const int   y1 = y0 + 1;
        const int   x1 = x0 + 1;
        const bool  vy0 = (y0 >= 0) && (y0 <= 63);
        const bool  vy1 = (y1 >= 0) && (y1 <= 63);
        const bool  vx0 = (x0 >= 0) && (x0 <= 63);
        const bool  vx1 = (x1 >= 0) && (x1 <= 63);
        const int   cy0 = y0 < 0 ? 0 : (y0 > 63 ? 63 : y0);
        const int   cy1 = y1 < 0 ? 0 : (y1 > 63 ? 63 : y1);
        const int   cx0 = x0 < 0 ? 0 : (x0 > 63 ? 63 : x0);
        const int   cx1 = x1 < 0 ? 0 : (x1 > 63 ? 63 : x1);
        s_idx[e * 4 + 0] = (unsigned short)(cy0 * 64 + cx0);
        s_idx[e * 4 + 1] = (unsigned short)(cy0 * 64 + cx1);
        s_idx[e * 4 + 2] = (unsigned short)(cy1 * 64 + cx0);
        s_idx[e * 4 + 3] = (unsigned short)(cy1 * 64 + cx1);
        s_wgt[e * 4 + 0] = (vy0 && vx0) ? (1.f - ly) * (1.f - lx) * mv : 0.f;
        s_wgt[e * 4 + 1] = (vy0 && vx1) ? (1.f - ly) * lx          * mv : 0.f;
        s_wgt[e * 4 + 2] = (vy1 && vx0) ? ly          * (1.f - lx) * mv : 0.f;
        s_wgt[e * 4 + 3] = (vy1 && vx1) ? ly          * lx          * mv : 0.f;
    }

    const int wave = t >> 5;
    const int lane = t & 31;
    const int lh   = lane & 15;
    const int hig  = (lane >> 4) & 1;  // lane half: K shifted by 2 in high half
    const int m0   = wave * 32;        // this wave owns Co rows [m0, m0+32)

    v8f acc[2][4] = {};                // 2 M-tiles x 4 N-tiles of 16x16

    const float* xb = x + (size_t)b * C_ * HW_;
    const float* wrow = weight + (size_t)t * KT_;          // row m = t
    const unsigned ldsA0 = (unsigned)(uintptr_t)&s_A[0][t * ASTR];
    const unsigned ldsA1 = (unsigned)(uintptr_t)&s_A[1][t * ASTR];

    // ---- stage one K-chunk (16 k-indices) into buffer `buf` ----
    auto stage = [&](int k0, int buf) {
        // A: async DMA of 16 contiguous weights for row m = t (4 x b128)
        const unsigned ldsA = buf ? ldsA1 : ldsA0;
#pragma unroll
        for (int i = 0; i < 4; ++i)
            async_lds_b128(ldsA + 16u * i, wrow + k0 + 4 * i);
        // B: 16x64 col tile via bilinear gathers (4 elements per thread),
        // stored k-pair interleaved so fragments are contiguous b64 loads.
#pragma unroll
        for (int e2 = 0; e2 < 4; ++e2) {
            const int ee   = e2 * 256 + t;     // 0..1023
            const int kk   = ee >> 6;          // 0..15
            const int q    = ee & 63;
            const int kidx = k0 + kk;
            const int c    = kidx / 9;
            const int tap  = kidx - c * 9;
            const float* xc = xb + c * HW_;
            const int te = (tap * 64 + q) * 4;
            const float v = s_wgt[te + 0] * xc[s_idx[te + 0]]
                          + s_wgt[te + 1] * xc[s_idx[te + 1]]
                          + s_wgt[te + 2] * xc[s_idx[te + 2]]
                          + s_wgt[te + 3] * xc[s_idx[te + 3]];
            s_B[buf][(kk >> 1) * BP_ + 2 * q + (kk & 1)] = v;
        }
    };

    // ---- WMMA over one staged chunk ----
    auto compute = [&](int buf) {
        const float* A  = s_A[buf];
        const float* Bm = s_B[buf];
#pragma unroll
        for (int ks = 0; ks < 16; ks += 4) {
            const int kA = ks + 2 * hig;       // even
            v2f af[2], bf[4];
#pragma unroll
            for (int mt = 0; mt < 2; ++mt) {
                const int m = m0 + mt * 16 + lh;
                af[mt] = *(const v2f*)(A + m * ASTR + kA);          // b64, 8B aligned
            }
#pragma unroll
            for (int nt = 0; nt < 4; ++nt) {
                const int n = nt * 16 + lh;
                bf[nt] = *(const v2f*)(Bm + (kA >> 1) * BP_ + 2 * n); // b64, 8B aligned
            }
#pragma unroll
            for (int mt = 0; mt < 2; ++mt)
#pragma unroll
                for (int nt = 0; nt < 4; ++nt)
                    acc[mt][nt] = __builtin_amdgcn_wmma_f32_16x16x4_f32(
                        false, af[mt], false, bf[nt],
                        (short)0, acc[mt][nt], false, false);
        }
    };

    // prologue: stage chunk 0
    stage(0, 0);
    wait_async0();
    __syncthreads();

    // main loop: stage chunk i+1 first, then compute chunk i under its latency
    for (int i = 0; i < NCHUNK; ++i) {
        const int cur = i & 1;
        if (i + 1 < NCHUNK)
            stage((i + 1) * 16, cur ^ 1);
        compute(cur);
        wait_async0();
        __syncthreads();
    }

    // ---------- epilogue: C/D layout -> out[b][co][p][q] + bias ----------
    float* ob = out + (size_t)b * CO_ * HW_ + p * 64;
#pragma unroll
    for (int mt = 0; mt < 2; ++mt) {
#pragma unroll
        for (int nt = 0; nt < 4; ++nt) {
            const int q   = nt * 16 + lh;
            const int coB = m0 + mt * 16 + 8 * hig;  // VGPR r: M = r (+8 for hi half)
#pragma unroll
            for (int r = 0; r < 8; ++r) {
                const int co = coB + r;
                ob[(size_t)co * HW_ + q] = acc[mt][nt][r] + bias[co];
            }
        }
    }
}

extern "C" void kernel_launch(void* const* d_in, const int* in_sizes, int n_in,
                              void* d_out, int out_size, void* d_ws, size_t ws_size,
                              hipStream_t stream) {
    const float* x      = (const float*)d_in[0];
    const float* offset = (const float*)d_in[1];
    const float* mask   = (const float*)d_in[2];
    const float* weight = (const float*)d_in[3];
    const float* bias   = (const float*)d_in[4];
    float* out = (float*)d_out;

    dim3 grid(B_ * 64);   // 8 batches x 64 output rows
    dim3 block(256);      // 8 waves
    hipLaunchKernelGGL(dcn_wmma_kernel, grid, block, 0, stream,
                       x, offset, mask, weight, bias, out);
}